// HistAttProd_18468359372839
// MI455X (gfx1250) — compile-verified
//
#include <hip/hip_runtime.h>
#include <hip/hip_bf16.h>
#include <math.h>

// Problem constants (match reference).
#define QDIM 128
#define KDIM 64
#define TLEN 200
#define HSTR 65          // padded LDS row stride (words) -> bank-conflict-free
#define NULL_ATT (-4096.0f)

typedef float v2f __attribute__((ext_vector_type(2)));
typedef float v8f __attribute__((ext_vector_type(8)));

// ---------------------------------------------------------------------------
// Kernel 1: q = x @ W.T + b  via V_WMMA_F32_16X16X4_F32.
// One wave computes one 16(M) x 16(N) tile of q; block = 4 waves covers N=64.
// A-frag layout (ISA 7.12.2, 32-bit A 16x4): lanes 0-15 hold K=k,k+1 in v0,v1;
// lanes 16-31 hold K=k+2,k+3. B-frag mirrored (4x16, rows striped over lanes).
// C/D: VGPR i -> M = i (lanes 0-15) / M = 8+i (lanes 16-31), N = lane%16.
// ---------------------------------------------------------------------------
__global__ void __launch_bounds__(128)
hist_att_proj_wmma(const float* __restrict__ X,    // [B, 128]
                   const float* __restrict__ W,    // [64, 128]
                   const float* __restrict__ bias, // [64]
                   float* __restrict__ Q)          // [B, 64]
{
    const int wave = threadIdx.x >> 5;      // 0..3  -> N tile
    const int lane = threadIdx.x & 31;
    const int half = lane >> 4;             // 0: K lo pair, 1: K hi pair
    const int l    = lane & 15;
    const int m0   = blockIdx.x * 16;
    const int n0   = wave * 16;

    const float* __restrict__ xr = X + (size_t)(m0 + l) * QDIM; // A row (M = l)
    const float* __restrict__ wr = W + (size_t)(n0 + l) * QDIM; // B col n: W[n][*]

    v8f c = {};
#pragma unroll
    for (int k = 0; k < QDIM; k += 4) {
        const int ka = k + half * 2;
        v2f a, bm;
        a.x  = xr[ka];
        a.y  = xr[ka + 1];
        bm.x = wr[ka];       // B[ka][n]   = W[n][ka]
        bm.y = wr[ka + 1];   // B[ka+1][n]
        c = __builtin_amdgcn_wmma_f32_16x16x4_f32(
                /*neg_a=*/false, a, /*neg_b=*/false, bm,
                /*c_mod=*/(short)0, c, /*reuse_a=*/false, /*reuse_b=*/false);
    }

    const int   n  = n0 + l;
    const float bn = bias[n];
#pragma unroll
    for (int i = 0; i < 8; ++i) {
        const int m = m0 + half * 8 + i;
        Q[(size_t)m * KDIM + n] = c[i] + bn;
    }
}

// ---------------------------------------------------------------------------
// Kernel 2: per-row masked-softmax attention. One 256-thread block per b.
// hist row slab (200x64 f32, 51.2KB) staged once in LDS (stride-65 pad),
// reused for the score pass and the weighted-sum pass -> single HBM pass.
// ---------------------------------------------------------------------------
__global__ void __launch_bounds__(256)
hist_att_softmax_reduce(const float* __restrict__ hist,  // [B, 200, 64]
                        const int*   __restrict__ hlen,  // [B]
                        const float* __restrict__ Q,     // [B, 64]
                        float* __restrict__ rep,         // [B, 64]
                        float* __restrict__ atten)       // [B, 200]
{
    __shared__ float hist_s[TLEN * HSTR];   // 52000 B, padded rows
    __shared__ float q_s[KDIM];
    __shared__ float sc_s[256];
    __shared__ float red_s[256];
    __shared__ float part_s[256];

    const int b   = blockIdx.x;
    const int tid = threadIdx.x;
    const float* __restrict__ hb = hist + (size_t)b * (TLEN * KDIM);

    // Coalesced float4 staging into padded LDS (k offsets stay 4-aligned).
    for (int i = tid; i < (TLEN * KDIM) / 4; i += 256) {
        const float4 v   = ((const float4*)hb)[i];
        const int    flat = i * 4;
        const int    t    = flat >> 6;          // /64
        const int    k    = flat & 63;
        float* dst = &hist_s[t * HSTR + k];
        dst[0] = v.x; dst[1] = v.y; dst[2] = v.z; dst[3] = v.w;
    }
    if (tid < KDIM) q_s[tid] = Q[(size_t)b * KDIM + tid];
    __syncthreads();

    // --- scores with length mask -------------------------------------------
    const int L = hlen[b];
    float sc = -INFINITY;
    if (tid < TLEN) {
        float dot = 0.0f;
#pragma unroll 16
        for (int k = 0; k < KDIM; ++k)
            dot = fmaf(q_s[k], hist_s[tid * HSTR + k], dot);
        sc = (tid < L) ? dot : NULL_ATT;
    }
    sc_s[tid]  = sc;
    red_s[tid] = sc;
    __syncthreads();

    // --- block max ----------------------------------------------------------
    for (int s = 128; s > 0; s >>= 1) {
        if (tid < s) red_s[tid] = fmaxf(red_s[tid], red_s[tid + s]);
        __syncthreads();
    }
    const float mx = red_s[0];
    __syncthreads();

    // --- exp + block sum ----------------------------------------------------
    const float e = (tid < TLEN) ? __expf(sc_s[tid] - mx) : 0.0f;
    red_s[tid] = e;
    __syncthreads();
    for (int s = 128; s > 0; s >>= 1) {
        if (tid < s) red_s[tid] += red_s[tid + s];
        __syncthreads();
    }
    const float w = e * (1.0f / red_s[0]);
    __syncthreads();
    sc_s[tid] = w;                         // weights now live in sc_s
    __syncthreads();

    if (tid < TLEN) atten[(size_t)b * TLEN + tid] = w;

    // --- weighted sum: rep[k] = sum_t w[t] * hist[t][k] ---------------------
    const int k    = tid & 63;
    const int part = tid >> 6;             // 0..3, 50 t-values each
    float acc = 0.0f;
    const int t0 = part * (TLEN / 4);
#pragma unroll 10
    for (int t = t0; t < t0 + TLEN / 4; ++t)
        acc = fmaf(sc_s[t], hist_s[t * HSTR + k], acc);
    part_s[tid] = acc;
    __syncthreads();
    if (tid < KDIM) {
        rep[(size_t)b * KDIM + tid] =
            part_s[tid] + part_s[tid + 64] + part_s[tid + 128] + part_s[tid + 192];
    }
}

// ---------------------------------------------------------------------------
extern "C" void kernel_launch(void* const* d_in, const int* in_sizes, int n_in,
                              void* d_out, int out_size, void* d_ws, size_t ws_size,
                              hipStream_t stream) {
    const float* x    = (const float*)d_in[0];   // [B,128]
    const float* hist = (const float*)d_in[1];   // [B,200,64]
    const int*   hlen = (const int*)  d_in[2];   // [B]
    const float* W    = (const float*)d_in[3];   // [64,128]
    const float* bias = (const float*)d_in[4];   // [64]

    const int B = in_sizes[0] / QDIM;            // 4096

    float* q     = (float*)d_ws;                 // [B,64] scratch (1 MB)
    float* rep   = (float*)d_out;                // [B,64]
    float* atten = (float*)d_out + (size_t)B * KDIM; // [B,200]

    hist_att_proj_wmma<<<dim3(B / 16), dim3(128), 0, stream>>>(x, W, bias, q);
    hist_att_softmax_reduce<<<dim3(B), dim3(256), 0, stream>>>(hist, hlen, q, rep, atten);
}